// UnifiedLossMemoryMultiFocalPercent_64776696759052
// MI455X (gfx1250) — compile-verified
//
#include <hip/hip_runtime.h>

typedef __attribute__((ext_vector_type(2))) float v2f;
typedef __attribute__((ext_vector_type(8))) float v8f;

#define GAMMA_F 16.0f
#define NEG_INF (-__builtin_inff())
#define LDA 268   // row stride in dwords: 16B-aligned rows (268*4=1072), bank-spread

// ---- (max, sumexp) pair merge, -inf safe --------------------------------

__device__ __forceinline__ void lse_merge(float& m1, float& s1, float m2, float s2) {
    float M = fmaxf(m1, m2);
    if (M == NEG_INF) { m1 = NEG_INF; s1 = 0.0f; return; }
    s1 = s1 * __expf(m1 - M) + s2 * __expf(m2 - M);
    m1 = M;
}

// ---- kernel 1: L2-normalize rows, fold gamma in, gather targets ---------

__global__ void normalize_kernel(const float* __restrict__ results,
                                 const int* __restrict__ labels,
                                 const int* __restrict__ indexes,
                                 float* __restrict__ xbuf,
                                 int* __restrict__ targets,
                                 int D) {
    const int row = blockIdx.x;
    const int tid = threadIdx.x;               // 64 threads, D=256 -> 64 float4
    const float4* src = (const float4*)(results + (size_t)row * D);
    float4 v = src[tid];
    float ss = v.x * v.x + v.y * v.y + v.z * v.z + v.w * v.w;
    #pragma unroll
    for (int off = 16; off > 0; off >>= 1) ss += __shfl_xor(ss, off);
    __shared__ float wsum[2];
    if ((tid & 31) == 0) wsum[tid >> 5] = ss;
    __syncthreads();
    float total = wsum[0] + wsum[1];
    float inv = GAMMA_F / fmaxf(sqrtf(total), 1e-12f);
    float4 o; o.x = v.x * inv; o.y = v.y * inv; o.z = v.z * inv; o.w = v.w * inv;
    ((float4*)(xbuf + (size_t)row * D))[tid] = o;
    if (tid == 0) targets[row] = labels[indexes[row]];
}

// ---- kernel 2: fp32 WMMA GEMM tile + fused masked two-pass LSE ----------
// Block = 256 threads (8 wave32). Tile = 128(M) x 128(N), full K=256 staged
// in ~268 KB LDS once. Waves arranged 4(M) x 2(N); each wave computes a
// 32x64 strip as 2x4 register block of 16x16 WMMA accumulators:
// 8 v_wmma per 6 ds-b64 fragment loads per K-step.

__global__ void __launch_bounds__(256)
gemm_lse_kernel(const float* __restrict__ xbuf,
                const float* __restrict__ features,
                const int* __restrict__ targets,
                const int* __restrict__ labels,
                float4* __restrict__ partials,
                int D) {
    extern __shared__ float smem[];
    float* As = smem;                  // 128 x LDA
    float* Bs = smem + 128 * LDA;      // 128 x LDA
    __shared__ __align__(16) int tgt_s[128];
    __shared__ __align__(16) int lab_s[128];
    __shared__ float redbuf[8][4];

    const int tid    = threadIdx.x;
    const int blockM = blockIdx.y * 128;
    const int blockN = blockIdx.x * 128;

    if (tid < 128) {
        tgt_s[tid] = targets[blockM + tid];
        lab_s[tid] = labels[blockN + tid];
    }

    // Stage A and B tiles: each 128 rows x 64 float4 = 8192 float4, 32/thread
    #pragma unroll 4
    for (int i = 0; i < 32; ++i) {
        int idx = tid + i * 256;
        int r = idx >> 6, c4 = idx & 63;
        float4 va = ((const float4*)(xbuf     + (size_t)(blockM + r) * D))[c4];
        float4 vb = ((const float4*)(features + (size_t)(blockN + r) * D))[c4];
        *(float4*)(As + r * LDA + c4 * 4) = va;
        *(float4*)(Bs + r * LDA + c4 * 4) = vb;
    }
    __syncthreads();

    const int lane = tid & 31;
    const int wave = tid >> 5;
    const int wm   = wave & 3;     // M strip: wm*32 (2 sub-tiles of 16)
    const int wn   = wave >> 2;    // N strip: wn*64 (4 sub-tiles of 16)
    const int q    = lane & 15;
    const int kh   = lane >> 4;    // K-half per fp32 WMMA A/B fragment layout

    const float* A0 = As + (wm * 32 +  0 + q) * LDA;
    const float* A1 = As + (wm * 32 + 16 + q) * LDA;
    const float* B0 = Bs + (wn * 64 +  0 + q) * LDA;
    const float* B1 = Bs + (wn * 64 + 16 + q) * LDA;
    const float* B2 = Bs + (wn * 64 + 32 + q) * LDA;
    const float* B3 = Bs + (wn * 64 + 48 + q) * LDA;

    v8f acc[2][4];
    #pragma unroll
    for (int t = 0; t < 2; ++t)
        #pragma unroll
        for (int j = 0; j < 4; ++j) acc[t][j] = (v8f){};

    #pragma unroll 4
    for (int k0 = 0; k0 < 256; k0 += 4) {
        const int kk = k0 + 2 * kh;            // lanes 0-15: K{0,1}; 16-31: K{2,3}
        v2f a0 = *(const v2f*)(A0 + kk);
        v2f a1 = *(const v2f*)(A1 + kk);
        v2f b0 = *(const v2f*)(B0 + kk);
        v2f b1 = *(const v2f*)(B1 + kk);
        v2f b2 = *(const v2f*)(B2 + kk);
        v2f b3 = *(const v2f*)(B3 + kk);
        acc[0][0] = __builtin_amdgcn_wmma_f32_16x16x4_f32(false, a0, false, b0, (short)0, acc[0][0], false, false);
        acc[0][1] = __builtin_amdgcn_wmma_f32_16x16x4_f32(false, a0, false, b1, (short)0, acc[0][1], false, false);
        acc[0][2] = __builtin_amdgcn_wmma_f32_16x16x4_f32(false, a0, false, b2, (short)0, acc[0][2], false, false);
        acc[0][3] = __builtin_amdgcn_wmma_f32_16x16x4_f32(false, a0, false, b3, (short)0, acc[0][3], false, false);
        acc[1][0] = __builtin_amdgcn_wmma_f32_16x16x4_f32(false, a1, false, b0, (short)0, acc[1][0], false, false);
        acc[1][1] = __builtin_amdgcn_wmma_f32_16x16x4_f32(false, a1, false, b1, (short)0, acc[1][1], false, false);
        acc[1][2] = __builtin_amdgcn_wmma_f32_16x16x4_f32(false, a1, false, b2, (short)0, acc[1][2], false, false);
        acc[1][3] = __builtin_amdgcn_wmma_f32_16x16x4_f32(false, a1, false, b3, (short)0, acc[1][3], false, false);
    }

    // ---- branchless two-pass masked LSE epilogue ----
    // C layout: element v of lane -> M = v + 8*kh, N = q (per 16x16 sub-tile).
    int labv[4];
    #pragma unroll
    for (int j = 0; j < 4; ++j) labv[j] = lab_s[wn * 64 + j * 16 + q];
    int4 tg0 = *(const int4*)(tgt_s + wm * 32 +  0 + 8 * kh);
    int4 tg1 = *(const int4*)(tgt_s + wm * 32 +  4 + 8 * kh - 4 * kh * 0); // t=0 rows
    int4 tg2 = *(const int4*)(tgt_s + wm * 32 + 16 + 8 * kh);
    int4 tg3 = *(const int4*)(tgt_s + wm * 32 + 20 + 8 * kh);
    // repair: tg1 should be rows +4..7 of t=0
    tg1 = *(const int4*)(tgt_s + wm * 32 + 4 + 8 * kh);
    int tgtv[2][8];
    tgtv[0][0]=tg0.x; tgtv[0][1]=tg0.y; tgtv[0][2]=tg0.z; tgtv[0][3]=tg0.w;
    tgtv[0][4]=tg1.x; tgtv[0][5]=tg1.y; tgtv[0][6]=tg1.z; tgtv[0][7]=tg1.w;
    tgtv[1][0]=tg2.x; tgtv[1][1]=tg2.y; tgtv[1][2]=tg2.z; tgtv[1][3]=tg2.w;
    tgtv[1][4]=tg3.x; tgtv[1][5]=tg3.y; tgtv[1][6]=tg3.z; tgtv[1][7]=tg3.w;

    float mp = NEG_INF, mn = NEG_INF;
    #pragma unroll
    for (int t = 0; t < 2; ++t)
        #pragma unroll
        for (int j = 0; j < 4; ++j)
            #pragma unroll
            for (int v = 0; v < 8; ++v) {
                float s = acc[t][j][v];
                bool pos = (tgtv[t][v] == labv[j]);
                mp = fmaxf(mp, pos ? -s : NEG_INF);
                mn = fmaxf(mn, pos ? NEG_INF : s);
            }

    float sp = 0.0f, sn = 0.0f;
    if (mp != NEG_INF) {           // rare: ~1/8000 element density of positives
        #pragma unroll
        for (int t = 0; t < 2; ++t)
            #pragma unroll
            for (int j = 0; j < 4; ++j)
                #pragma unroll
                for (int v = 0; v < 8; ++v) {
                    float s = acc[t][j][v];
                    bool pos = (tgtv[t][v] == labv[j]);
                    sp += __expf((pos ? -s : NEG_INF) - mp);   // exp(-inf)=0
                }
    }
    if (mn != NEG_INF) {
        #pragma unroll
        for (int t = 0; t < 2; ++t)
            #pragma unroll
            for (int j = 0; j < 4; ++j)
                #pragma unroll
                for (int v = 0; v < 8; ++v) {
                    float s = acc[t][j][v];
                    bool pos = (tgtv[t][v] == labv[j]);
                    sn += __expf((pos ? NEG_INF : s) - mn);    // exp(-inf)=0
                }
    }

    // Wave32 reduction of (max, sumexp) pairs
    #pragma unroll
    for (int off = 16; off > 0; off >>= 1) {
        float m2 = __shfl_xor(mp, off), s2 = __shfl_xor(sp, off);
        lse_merge(mp, sp, m2, s2);
        m2 = __shfl_xor(mn, off); s2 = __shfl_xor(sn, off);
        lse_merge(mn, sn, m2, s2);
    }
    if (lane == 0) {
        redbuf[wave][0] = mp; redbuf[wave][1] = sp;
        redbuf[wave][2] = mn; redbuf[wave][3] = sn;
    }
    __syncthreads();
    if (tid == 0) {
        float Mp = NEG_INF, Sp = 0.0f, Mn = NEG_INF, Sn = 0.0f;
        #pragma unroll
        for (int w = 0; w < 8; ++w) {
            lse_merge(Mp, Sp, redbuf[w][0], redbuf[w][1]);
            lse_merge(Mn, Sn, redbuf[w][2], redbuf[w][3]);
        }
        partials[blockIdx.y * gridDim.x + blockIdx.x] = make_float4(Mp, Sp, Mn, Sn);
    }
}

// ---- kernel 3: merge partials, softplus ---------------------------------

__global__ void finalize_kernel(const float4* __restrict__ partials, int nparts,
                                float* __restrict__ out) {
    const int tid = threadIdx.x;   // 256 threads
    float mp = NEG_INF, sp = 0.0f, mn = NEG_INF, sn = 0.0f;
    for (int i = tid; i < nparts; i += 256) {
        float4 p = partials[i];
        lse_merge(mp, sp, p.x, p.y);
        lse_merge(mn, sn, p.z, p.w);
    }
    #pragma unroll
    for (int off = 16; off > 0; off >>= 1) {
        float m2 = __shfl_xor(mp, off), s2 = __shfl_xor(sp, off);
        lse_merge(mp, sp, m2, s2);
        m2 = __shfl_xor(mn, off); s2 = __shfl_xor(sn, off);
        lse_merge(mn, sn, m2, s2);
    }
    __shared__ float red[8][4];
    const int lane = tid & 31, wave = tid >> 5;
    if (lane == 0) { red[wave][0] = mp; red[wave][1] = sp; red[wave][2] = mn; red[wave][3] = sn; }
    __syncthreads();
    if (tid == 0) {
        float Mp = NEG_INF, Sp = 0.0f, Mn = NEG_INF, Sn = 0.0f;
        #pragma unroll
        for (int w = 0; w < 8; ++w) {
            lse_merge(Mp, Sp, red[w][0], red[w][1]);
            lse_merge(Mn, Sn, red[w][2], red[w][3]);
        }
        float lse_p = Mp + logf(Sp);
        float lse_n = Mn + logf(Sn);
        float z = lse_p + lse_n;
        float loss = (z > 0.0f) ? (z + log1pf(__expf(-z))) : log1pf(__expf(z));
        out[0] = loss;
    }
}

// ---- launcher ------------------------------------------------------------

extern "C" void kernel_launch(void* const* d_in, const int* in_sizes, int n_in,
                              void* d_out, int out_size, void* d_ws, size_t ws_size,
                              hipStream_t stream) {
    const float* results  = (const float*)d_in[0];
    const float* features = (const float*)d_in[1];
    const int*   labels   = (const int*)d_in[2];
    const int*   indexes  = (const int*)d_in[3];

    const int B = in_sizes[3];            // 2048
    const int N = in_sizes[2];            // 32768
    const int D = in_sizes[0] / B;        // 256

    // Workspace layout
    float* xbuf = (float*)d_ws;
    size_t off  = (size_t)B * D * sizeof(float);
    int* targets = (int*)((char*)d_ws + off);
    off += (size_t)B * sizeof(int);
    off = (off + 255) & ~(size_t)255;
    float4* partials = (float4*)((char*)d_ws + off);

    // 1) normalize + gamma fold + target gather
    normalize_kernel<<<B, D / 4, 0, stream>>>(results, labels, indexes, xbuf, targets, D);

    // 2) WMMA GEMM + fused masked two-pass logsumexp
    dim3 grid(N / 128, B / 128);
    size_t smem = (size_t)(128 + 128) * LDA * sizeof(float);   // ~268 KB of 320 KB WGP LDS
    gemm_lse_kernel<<<grid, 256, smem, stream>>>(xbuf, features, targets, labels, partials, D);

    // 3) global merge + softplus
    int nparts = (N / 128) * (B / 128);
    finalize_kernel<<<1, 256, 0, stream>>>(partials, nparts, (float*)d_out);
}